// CrossAttentionSelector_73478300500454
// MI455X (gfx1250) — compile-verified
//
#include <hip/hip_runtime.h>
#include <math.h>

// Problem constants (from reference): B=16, N=4096, D=1024, H=16, head=64
#define BB 16
#define NN 4096
#define DD 1024
#define HH 16
#define DHD 64
#define SCALE 0.125f   // 64^-0.5
#define KSPLIT 4
#define YP (BB * HH * DD)   // one y-partial = 262144 floats

typedef float v2f __attribute__((ext_vector_type(2)));
typedef float v4f __attribute__((ext_vector_type(4)));
typedef float v8f __attribute__((ext_vector_type(8)));

// ---------------- K1: Q = state @ Wq^T + bq   [B, D] ----------------
__global__ void k_qproj(const float* __restrict__ state, const float* __restrict__ Wq,
                        const float* __restrict__ bq, float* __restrict__ Q) {
    int t = blockIdx.x * blockDim.x + threadIdx.x;  // B*D threads
    int b = t >> 10, j = t & 1023;
    const v4f* s = (const v4f*)(state + b * DD);
    const v4f* w = (const v4f*)(Wq + (size_t)j * DD);
    float acc = bq[j];
#pragma unroll 4
    for (int m = 0; m < DD / 4; ++m) {
        v4f a = s[m], ww = w[m];
        acc += a.x * ww.x + a.y * ww.y + a.z * ww.z + a.w * ww.w;
    }
    Q[t] = acc;
}

// ---------------- K2: q~[b,h,m] = scale * sum_d Q[b,h,d]*Wk[h*64+d, m] ----------------
__global__ void k_qtil(const float* __restrict__ Q, const float* __restrict__ Wk,
                       float* __restrict__ qt) {
    int t = blockIdx.x * blockDim.x + threadIdx.x;  // B*H*D threads
    int b = t / (HH * DD);
    int r = t % (HH * DD);
    int h = r >> 10, m = r & 1023;
    const float* qrow = Q + b * DD + h * DHD;
    float acc = 0.f;
#pragma unroll 8
    for (int d = 0; d < DHD; ++d)
        acc += qrow[d] * Wk[(size_t)(h * DHD + d) * DD + m];
    qt[t] = acc * SCALE;
}

// ---------------- K2b: c[b,h] = scale * Q[b,h,:]·bk_head ----------------
__global__ void k_cconst(const float* __restrict__ Q, const float* __restrict__ bk,
                         float* __restrict__ cc) {
    int t = threadIdx.x;  // 256 = B*H
    int b = t >> 4, h = t & 15;
    float acc = 0.f;
#pragma unroll 8
    for (int d = 0; d < DHD; ++d)
        acc += Q[b * DD + h * DHD + d] * bk[h * DHD + d];
    cc[t] = acc * SCALE;
}

// ---------------- K3 (WMMA pass 1): scores[b,n,h] = q~·x + c ; also head-mean ----------
// 1 wave per block; each wave: 16 candidate rows x 16 heads, K-loop over D=1024 (step 4).
__global__ void k_scores(const float* __restrict__ cand, const float* __restrict__ qt,
                         const float* __restrict__ cc, float* __restrict__ sc,
                         float* __restrict__ mean) {
    int blk = blockIdx.x;          // b * (N/16) + tile
    int b = blk >> 8;              // N/16 = 256
    int tile = blk & 255;
    int lane = threadIdx.x;
    int half = lane >> 4, li = lane & 15;
    // A frag: M = candidate row (li), K pair at 2*half  -> contiguous float2
    const float* ap = cand + ((size_t)b * NN + tile * 16 + li) * DD + 2 * half;
    // B frag: col = head (li), K pair at 2*half -> contiguous float2 of q~[b][h][m]
    const float* bp = qt + ((size_t)b * HH + li) * DD + 2 * half;
    v8f c = {};
#pragma unroll 8
    for (int k = 0; k < DD; k += 4) {
        v2f a = *(const v2f*)(ap + k);
        v2f bb = *(const v2f*)(bp + k);
        c = __builtin_amdgcn_wmma_f32_16x16x4_f32(false, a, false, bb, (short)0, c,
                                                  false, false);
    }
    float cadd = cc[b * HH + li];  // per-head constant
    int nbase = tile * 16 + 8 * half;
#pragma unroll
    for (int i = 0; i < 8; ++i) {
        int n = nbase + i;  // C: VGPR i -> rows i / i+8
        float v = c[i] + cadd;
        sc[(size_t)(b * NN + n) * HH + li] = v;
        // head-mean: reduce across the 16 lanes of this half (heads)
        float s = v;
        s += __shfl_xor(s, 1, 32);
        s += __shfl_xor(s, 2, 32);
        s += __shfl_xor(s, 4, 32);
        s += __shfl_xor(s, 8, 32);
        if (li == 0) mean[b * NN + n] = s * (1.0f / HH);
    }
}

// ------- K4a: per-(b,h) max & sum(exp) over N, then convert scores->attn in place -----
__global__ void k_rowsoft(float* __restrict__ sc) {
    __shared__ float red[128];
    int bh = blockIdx.x;
    int b = bh >> 4, h = bh & 15;
    int tid = threadIdx.x;
    float* p = sc + (size_t)b * NN * HH + h;
    float mx = -INFINITY;
    for (int n = tid; n < NN; n += 128) mx = fmaxf(mx, p[(size_t)n * HH]);
    red[tid] = mx; __syncthreads();
    for (int s = 64; s > 0; s >>= 1) {
        if (tid < s) red[tid] = fmaxf(red[tid], red[tid + s]);
        __syncthreads();
    }
    mx = red[0]; __syncthreads();
    float sm = 0.f;
    for (int n = tid; n < NN; n += 128) sm += expf(p[(size_t)n * HH] - mx);
    red[tid] = sm; __syncthreads();
    for (int s = 64; s > 0; s >>= 1) {
        if (tid < s) red[tid] += red[tid + s];
        __syncthreads();
    }
    float sum = red[0]; __syncthreads();
    for (int n = tid; n < NN; n += 128)
        p[(size_t)n * HH] = expf(p[(size_t)n * HH] - mx) / sum;
}

// ---- K4c: per-b softmax over mean scores -> attn_weights output + top-8 ----
__global__ void k_softtop(const float* __restrict__ mean, float* __restrict__ out_attn,
                          float* __restrict__ out_idx, float* __restrict__ out_val) {
    __shared__ float red[256];
    __shared__ int redi[256];
    __shared__ int sel[8];
    __shared__ float s_bmax, s_bsum;
    int b = blockIdx.x, tid = threadIdx.x;
    const float* p = mean + b * NN;
    float mx = -INFINITY;
    for (int n = tid; n < NN; n += 256) mx = fmaxf(mx, p[n]);
    red[tid] = mx; __syncthreads();
    for (int s = 128; s > 0; s >>= 1) {
        if (tid < s) red[tid] = fmaxf(red[tid], red[tid + s]);
        __syncthreads();
    }
    mx = red[0]; __syncthreads();
    float sm = 0.f;
    for (int n = tid; n < NN; n += 256) sm += expf(p[n] - mx);
    red[tid] = sm; __syncthreads();
    for (int s = 128; s > 0; s >>= 1) {
        if (tid < s) red[tid] += red[tid + s];
        __syncthreads();
    }
    if (tid == 0) { s_bmax = mx; s_bsum = red[0]; }
    __syncthreads();
    float bmax = s_bmax, bsum = s_bsum;
    for (int n = tid; n < NN; n += 256)
        out_attn[b * NN + n] = expf(p[n] - bmax) / bsum;
    // top-8, JAX tie-break: equal value -> lower index
    for (int it = 0; it < 8; ++it) {
        float bestv = -INFINITY;
        int besti = NN;
        for (int n = tid; n < NN; n += 256) {
            bool skip = false;
            for (int j = 0; j < it; ++j)
                if (sel[j] == n) skip = true;
            float v = p[n];
            if (!skip && (v > bestv || (v == bestv && n < besti))) { bestv = v; besti = n; }
        }
        red[tid] = bestv; redi[tid] = besti; __syncthreads();
        for (int s = 128; s > 0; s >>= 1) {
            if (tid < s) {
                float v2 = red[tid + s]; int i2 = redi[tid + s];
                if (v2 > red[tid] || (v2 == red[tid] && i2 < redi[tid])) {
                    red[tid] = v2; redi[tid] = i2;
                }
            }
            __syncthreads();
        }
        if (tid == 0) {
            sel[it] = redi[0];
            out_idx[b * 8 + it] = (float)redi[0];
            out_val[b * 8 + it] = expf(red[0] - bmax) / bsum;
        }
        __syncthreads();
    }
}

// ---------------- K5 (WMMA pass 2, split-K x4): y_s[b,h,:] = sum_n attn * x ----------
// 1 wave per block: M = 16 heads, N = 16 D-columns, K = 1024 candidate rows per split.
__global__ void k_wsum(const float* __restrict__ cand, const float* __restrict__ attn,
                       float* __restrict__ ypart) {
    int blk = blockIdx.x;          // b*256 + dt*4 + s
    int b = blk >> 8;
    int r = blk & 255;
    int dt = r >> 2;
    int s = r & 3;
    int lane = threadIdx.x;
    int half = lane >> 4, li = lane & 15;
    int n0 = s * (NN / KSPLIT);
    // A frag: M = head (li), K pair at n = n0 + k + 2*half  (attn stored [b][n][h])
    const float* ap = attn + ((size_t)(b * NN + n0 + 2 * half)) * HH + li;
    // B frag: col = dt*16 + li, K rows n and n+1
    const float* bp = cand + ((size_t)(b * NN + n0 + 2 * half)) * DD + dt * 16 + li;
    v8f c = {};
#pragma unroll 8
    for (int k = 0; k < NN / KSPLIT; k += 4) {
        v2f a, bb;
        a.x = ap[(size_t)k * HH];
        a.y = ap[(size_t)k * HH + HH];
        bb.x = bp[(size_t)k * DD];
        bb.y = bp[(size_t)k * DD + DD];
        c = __builtin_amdgcn_wmma_f32_16x16x4_f32(false, a, false, bb, (short)0, c,
                                                  false, false);
    }
    float* yo = ypart + (size_t)s * YP;
#pragma unroll
    for (int i = 0; i < 8; ++i) {
        int hr = i + 8 * half;
        yo[((size_t)b * HH + hr) * DD + dt * 16 + li] = c[i];
    }
}

// ------- K6: ctx[b, h*64+d] = bv + Wv_row · (y0+y1+y2+y3)[b,h,:]  (folds split-K) ------
__global__ void k_ctx(const float* __restrict__ ypart, const float* __restrict__ Wv,
                      const float* __restrict__ bv, float* __restrict__ ctx) {
    int t = blockIdx.x * blockDim.x + threadIdx.x;  // B*D
    int b = t >> 10, j = t & 1023, h = j >> 6;
    size_t row = ((size_t)b * HH + h) * DD;
    const v4f* w  = (const v4f*)(Wv + (size_t)j * DD);
    const v4f* y0 = (const v4f*)(ypart + row);
    const v4f* y1 = (const v4f*)(ypart + YP + row);
    const v4f* y2 = (const v4f*)(ypart + 2 * (size_t)YP + row);
    const v4f* y3 = (const v4f*)(ypart + 3 * (size_t)YP + row);
    float acc = bv[j];
#pragma unroll 4
    for (int m = 0; m < DD / 4; ++m) {
        v4f a = y0[m] + y1[m] + y2[m] + y3[m];
        v4f ww = w[m];
        acc += a.x * ww.x + a.y * ww.y + a.z * ww.z + a.w * ww.w;
    }
    ctx[t] = acc;
}

// ---------------- K7: selected_info = ctx @ Wo^T + bo ----------------
__global__ void k_oproj(const float* __restrict__ ctx, const float* __restrict__ Wo,
                        const float* __restrict__ bo, float* __restrict__ out) {
    int t = blockIdx.x * blockDim.x + threadIdx.x;  // B*D
    int b = t >> 10, i = t & 1023;
    const v4f* w = (const v4f*)(Wo + (size_t)i * DD);
    const v4f* cx = (const v4f*)(ctx + b * DD);
    float acc = bo[i];
#pragma unroll 4
    for (int m = 0; m < DD / 4; ++m) {
        v4f a = cx[m], ww = w[m];
        acc += a.x * ww.x + a.y * ww.y + a.z * ww.z + a.w * ww.w;
    }
    out[t] = acc;
}

extern "C" void kernel_launch(void* const* d_in, const int* in_sizes, int n_in,
                              void* d_out, int out_size, void* d_ws, size_t ws_size,
                              hipStream_t stream) {
    const float* state = (const float*)d_in[0];
    const float* cand  = (const float*)d_in[1];
    const float* Wq    = (const float*)d_in[2];
    const float* bq    = (const float*)d_in[3];
    const float* Wk    = (const float*)d_in[4];
    const float* bk    = (const float*)d_in[5];
    const float* Wv    = (const float*)d_in[6];
    const float* bv    = (const float*)d_in[7];
    const float* Wo    = (const float*)d_in[8];
    const float* bo    = (const float*)d_in[9];

    float* ws    = (float*)d_ws;
    float* Q     = ws;                            // B*D     = 16384
    float* qt    = Q + BB * DD;                   // B*H*D   = 262144
    float* cc    = qt + BB * HH * DD;             // B*H     = 256
    float* sc    = cc + BB * HH;                  // B*N*H   = 1048576
    float* mean  = sc + (size_t)BB * NN * HH;     // B*N     = 65536
    float* ypart = mean + BB * NN;                // 4*B*H*D = 1048576
    float* ctx   = ypart + (size_t)KSPLIT * YP;   // B*D     = 16384

    float* out      = (float*)d_out;
    float* out_sel  = out;                      // [B, D]   16384
    float* out_attn = out + BB * DD;            // [B, N]   65536
    float* out_idx  = out_attn + BB * NN;       // [B, 8]   128 (indices as float)
    float* out_val  = out_idx + BB * 8;         // [B, 8]   128

    k_qproj  <<<BB * DD / 256, 256, 0, stream>>>(state, Wq, bq, Q);
    k_qtil   <<<BB * HH * DD / 256, 256, 0, stream>>>(Q, Wk, qt);
    k_cconst <<<1, 256, 0, stream>>>(Q, bk, cc);
    k_scores <<<BB * (NN / 16), 32, 0, stream>>>(cand, qt, cc, sc, mean);
    k_rowsoft<<<BB * HH, 128, 0, stream>>>(sc);
    k_softtop<<<BB, 256, 0, stream>>>(mean, out_attn, out_idx, out_val);
    k_wsum   <<<BB * (DD / 16) * KSPLIT, 32, 0, stream>>>(cand, sc, ypart);
    k_ctx    <<<BB * DD / 256, 256, 0, stream>>>(ypart, Wv, bv, ctx);
    k_oproj  <<<BB * DD / 256, 256, 0, stream>>>(ctx, Wo, bo, out_sel);
}